// PosFusionEmbedding_36395552866537
// MI455X (gfx1250) — compile-verified
//
#include <hip/hip_runtime.h>

// PosFusionEmbedding, MI455X/gfx1250.
//
// Separable rewrite:  out[b,l,m,:] = relu( Tss[iss] + Tee[iee] )
//   Tss = pe_ss @ W[:H] + b    (1025 x 160, f32)
//   Tee = pe_ee @ W[H:]        (1025 x 160, f32)
// Kernel 1: two tiny GEMMs via V_WMMA_F32_16X16X4_F32 (full f32 precision),
//           one wave per 16x16 tile, K-loop fully unrolled (40 wmma).
// Kernel 2: memory-bound gather+add+relu; 168 MB of NT output writes dominate
//           (~7.2 us at 23.3 TB/s HBM); tables stay hot in L2/WGP$.

#define MAXSEQ 512
#define TBL    1025      // 2*MAXSEQ + 1
#define H      160
#define B_     4
#define L_     256

typedef __attribute__((ext_vector_type(2))) float v2f;
typedef __attribute__((ext_vector_type(4))) float v4f;
typedef __attribute__((ext_vector_type(8))) float v8f;

// ---------------------------------------------------------------------------
// Kernel 1: T[t][i][h] = sum_c pe_t[i][c] * W[t*H + c][h]   (+ bias for t==0)
// Tiles: 2 tables * ceil(1025/16)=65 Mtiles * 160/16=10 Ntiles = 1300 waves.
// Edge handling: A-row is CLAMPED (not zeroed) -- garbage rows of A only
// produce garbage in the same D rows, which the store guard discards. This
// keeps the inner loop free of EXEC masking.
// ---------------------------------------------------------------------------
__global__ __launch_bounds__(32) void build_tables_kernel(
    const float* __restrict__ pe_ss, const float* __restrict__ pe_ee,
    const float* __restrict__ W,     const float* __restrict__ bias,
    float* __restrict__ T)           // [2][TBL][H] in workspace
{
    const int lane = threadIdx.x;            // wave32
    const int NTN  = H / 16;                 // 10
    const int NTM  = (TBL + 15) / 16;        // 65

    const int tile = blockIdx.x;             // 0..1299
    const int t    = tile / (NTM * NTN);     // which table (0=ss, 1=ee)
    const int r    = tile - t * (NTM * NTN);
    const int mt   = r / NTN;
    const int nt   = r - mt * NTN;

    const float* __restrict__ pe = (t == 0) ? pe_ss : pe_ee;
    const int i0    = mt * 16;               // table-row tile origin (M)
    const int h0    = nt * 16;               // output-col tile origin (N)
    const int wrow0 = t * H;                 // row offset into W [2H, H]

    const int lo = lane & 15;                // M row / N col within tile
    const int hi = lane >> 4;                // 0: K={0,1}  1: K={2,3}

    // A source row, clamped at the 1025 edge (see note above).
    const int arow = (i0 + lo < TBL) ? (i0 + lo) : (TBL - 1);
    const float* __restrict__ aptr = pe + (size_t)arow * H;
    const float* __restrict__ bptr = W + (size_t)wrow0 * H + h0 + lo;

    v8f acc = {};
    #pragma unroll
    for (int k0 = 0; k0 < H; k0 += 4) {
        const int kb = k0 + 2 * hi;          // this lane's K base (even)

        // A 16x4 f32: v0 = K {0|2}, v1 = K {1|3}  (contiguous pair, one b64)
        const v2f a = *(const v2f*)(aptr + kb);

        // B 4x16 f32: lane = column N, v0/v1 = consecutive K (stride-H apart)
        v2f b;
        b.x = bptr[(size_t)kb * H];
        b.y = bptr[(size_t)(kb + 1) * H];

        // D = A*B + C   (8 args: neg_a, A, neg_b, B, c_mod, C, reuse_a, reuse_b)
        acc = __builtin_amdgcn_wmma_f32_16x16x4_f32(
            false, a, false, b, (short)0, acc, false, false);
    }

    // Fold bias into the ss table only (so kernel 2 is a pure 2-way add).
    const float bv = (t == 0) ? bias[h0 + lo] : 0.0f;

    // C/D layout: VGPR v -> M = v + 8*hi,  N = lo
    float* __restrict__ Tt = T + (size_t)t * TBL * H + (size_t)i0 * H + h0 + lo;
    if (i0 + 16 <= TBL) {                    // 64 of 65 M-tiles: no guard
        #pragma unroll
        for (int v = 0; v < 8; ++v)
            Tt[(size_t)(v + 8 * hi) * H] = acc[v] + bv;
    } else {                                 // last tile: only row 1024 valid
        #pragma unroll
        for (int v = 0; v < 8; ++v)
            if (i0 + v + 8 * hi < TBL)
                Tt[(size_t)(v + 8 * hi) * H] = acc[v] + bv;
    }
}

// ---------------------------------------------------------------------------
// Kernel 2: out[b,l,m,h] = relu(Tss[ds+512][h] + Tee[de+512][h])
// One float4 per thread; L*L = 65536 so pair decode is shifts/masks.
// Purely store-bandwidth bound: 168 MB of coalesced non-temporal writes.
// ---------------------------------------------------------------------------
__global__ __launch_bounds__(256) void fuse_relu_kernel(
    const int* __restrict__ pos_s, const int* __restrict__ pos_e,
    const float* __restrict__ T, v4f* __restrict__ out)
{
    const int gid  = blockIdx.x * 256 + threadIdx.x;
    const int Q    = H / 4;                  // 40 float4 per (b,l,m)
    const int pair = gid / Q;
    const int hq   = gid - pair * Q;

    const int b  = pair >> 16;               // L*L == 65536
    const int lm = pair & 0xFFFF;
    const int l  = lm >> 8;
    const int m  = lm & 0xFF;

    const int base = b * L_;
    const int iss  = pos_s[base + l] - pos_s[base + m] + MAXSEQ;  // [257,767]
    const int iee  = pos_e[base + l] - pos_e[base + m] + MAXSEQ;

    const v4f a = *((const v4f*)(T + (size_t)iss * H) + hq);
    const v4f c = *((const v4f*)(T + (size_t)(TBL + iee) * H) + hq);

    v4f o;
    o.x = fmaxf(a.x + c.x, 0.0f);
    o.y = fmaxf(a.y + c.y, 0.0f);
    o.z = fmaxf(a.z + c.z, 0.0f);
    o.w = fmaxf(a.w + c.w, 0.0f);
    __builtin_nontemporal_store(o, out + gid);   // stream out, don't cache
}

extern "C" void kernel_launch(void* const* d_in, const int* in_sizes, int n_in,
                              void* d_out, int out_size, void* d_ws, size_t ws_size,
                              hipStream_t stream) {
    // setup_inputs() order: pos_s, pos_e, pe_ss, pe_ee, W, b
    const int*   pos_s = (const int*)  d_in[0];
    const int*   pos_e = (const int*)  d_in[1];
    const float* pe_ss = (const float*)d_in[2];
    const float* pe_ee = (const float*)d_in[3];
    const float* W     = (const float*)d_in[4];
    const float* bias  = (const float*)d_in[5];

    float* T = (float*)d_ws;                 // [2][1025][160] f32 = 1.31 MB

    // Table precompute: 1300 wmma-tiles, one wave each.
    const int n_tiles = 2 * ((TBL + 15) / 16) * (H / 16);
    build_tables_kernel<<<n_tiles, 32, 0, stream>>>(pe_ss, pe_ee, W, bias, T);

    // Fusion: B*L*L*H/4 float4 = 10,485,760 threads.
    const int total_v4 = B_ * L_ * L_ * (H / 4);
    fuse_relu_kernel<<<total_v4 / 256, 256, 0, stream>>>(pos_s, pos_e, T,
                                                         (v4f*)d_out);
}